// SparseConv3d_16527034155710
// MI455X (gfx1250) — compile-verified
//
#include <hip/hip_runtime.h>
#include <hip/hip_bf16.h>

typedef __attribute__((ext_vector_type(2))) float v2f;
typedef __attribute__((ext_vector_type(8))) float v8f;

#define KOFF 27   // 3x3x3 offsets
#define CI   32
#define CO   32

// One wave32 computes a 32-voxel x 32-Co output tile (2 M-tiles x 2 Co-tiles of
// 16x16) with V_WMMA_F32_16X16X4_F32. Register blocking on M doubles WMMAs per
// B-fragment load.
// A fragment (16x4 f32): lane<16 -> M=lane, K={0,1}; lane>=16 -> M=lane-16, K={2,3}.
// B fragment (4x16 f32): (K = 2*(lane>=16)+c, N = lane&15).
// C/D (16x16 f32): VGPR v -> M = v + 8*(lane>=16), col = lane&15.
__global__ __launch_bounds__(256) void SparseConv3d_wmma_kernel(
    const float* __restrict__ feats,   // [N, CI]
    const float* __restrict__ weight,  // [CO, KOFF, CI]  (Co,Kd,Kh,Kw,Ci flattened)
    const float* __restrict__ bias,    // [CO]
    const int*   __restrict__ nidx,    // [KOFF, N], -1 = missing neighbor
    float* __restrict__ out,           // [N, CO]
    int N)
{
    const int lane = threadIdx.x & 31;
    const int wave = threadIdx.x >> 5;
    const int tile = blockIdx.x * 8 + wave;          // 32-voxel tile
    const int ntiles = (N + 31) >> 5;
    if (tile >= ntiles) return;                      // wave-uniform: EXEC stays all-1s

    const int lo = lane & 15;                        // column / M-row selector
    const int hi = lane >> 4;                        // K-half selector (0 or 1)
    const int base = tile * 32;
    const int m0 = base + lo;                        // voxel for M-tile 0
    const int m1 = base + 16 + lo;                   // voxel for M-tile 1
    const int m0c = (m0 < N) ? m0 : (N - 1);
    const int m1c = (m1 < N) ? m1 : (N - 1);

    v8f acc00 = {};                                  // M-tile 0, co 0..15
    v8f acc01 = {};                                  // M-tile 0, co 16..31
    v8f acc10 = {};                                  // M-tile 1, co 0..15
    v8f acc11 = {};                                  // M-tile 1, co 16..31

    // Per-lane base into weight for its output channel (co = lo and lo+16).
    const float* __restrict__ wco0 = weight + (size_t)(lo +  0) * (KOFF * CI);
    const float* __restrict__ wco1 = weight + (size_t)(lo + 16) * (KOFF * CI);

    for (int k = 0; k < KOFF; ++k) {
        const int r0 = nidx[(size_t)k * N + m0c];
        const int r1 = nidx[(size_t)k * N + m1c];
        const bool v0 = (r0 >= 0) & (m0 < N);
        const bool v1 = (r1 >= 0) & (m1 < N);
        const float mk0 = v0 ? 1.0f : 0.0f;
        const float mk1 = v1 ? 1.0f : 0.0f;

        const float* __restrict__ arow0 = feats + (size_t)(v0 ? r0 : 0) * CI;
        const float* __restrict__ arow1 = feats + (size_t)(v1 ? r1 : 0) * CI;
        const float* __restrict__ wk0   = wco0 + k * CI;
        const float* __restrict__ wk1   = wco1 + k * CI;

        #pragma unroll
        for (int j = 0; j < 8; ++j) {                // Ci = 32 -> 8 K-steps of 4
            const int ci = j * 4 + hi * 2;           // this lane's 2 consecutive channels
            v2f a0 = *(const v2f*)(arow0 + ci);      // global_load_b64 (8B aligned)
            v2f a1 = *(const v2f*)(arow1 + ci);
            a0.x *= mk0; a0.y *= mk0;                // zero missing neighbors, EXEC unchanged
            a1.x *= mk1; a1.y *= mk1;
            const v2f b0 = *(const v2f*)(wk0 + ci);  // B frag, co tile 0
            const v2f b1 = *(const v2f*)(wk1 + ci);  // B frag, co tile 1
            acc00 = __builtin_amdgcn_wmma_f32_16x16x4_f32(
                        false, a0, false, b0, (short)0, acc00, false, false);
            acc10 = __builtin_amdgcn_wmma_f32_16x16x4_f32(
                        false, a1, false, b0, (short)0, acc10, false, false);
            acc01 = __builtin_amdgcn_wmma_f32_16x16x4_f32(
                        false, a0, false, b1, (short)0, acc01, false, false);
            acc11 = __builtin_amdgcn_wmma_f32_16x16x4_f32(
                        false, a1, false, b1, (short)0, acc11, false, false);
        }
    }

    const float bv0 = bias[lo];
    const float bv1 = bias[lo + 16];

    #pragma unroll
    for (int v = 0; v < 8; ++v) {
        const int row0 = base + v + hi * 8;          // D layout: M = v + 8*hi
        const int row1 = base + 16 + v + hi * 8;
        if (row0 < N) {
            out[(size_t)row0 * CO + lo]      = acc00[v] + bv0;
            out[(size_t)row0 * CO + lo + 16] = acc01[v] + bv1;
        }
        if (row1 < N) {
            out[(size_t)row1 * CO + lo]      = acc10[v] + bv0;
            out[(size_t)row1 * CO + lo + 16] = acc11[v] + bv1;
        }
    }
}

extern "C" void kernel_launch(void* const* d_in, const int* in_sizes, int n_in,
                              void* d_out, int out_size, void* d_ws, size_t ws_size,
                              hipStream_t stream) {
    const float* feats  = (const float*)d_in[0];   // [N, 32]
    const float* weight = (const float*)d_in[1];   // [32, 27, 32]
    const float* bias   = (const float*)d_in[2];   // [32]
    const int*   nidx   = (const int*)d_in[3];     // [27, N]
    float* out = (float*)d_out;                    // [N, 32]

    const int N = in_sizes[0] / CI;
    const int ntiles = (N + 31) / 32;
    const int blocks = (ntiles + 7) / 8;           // 8 waves (tiles) per 256-thread block

    SparseConv3d_wmma_kernel<<<blocks, 256, 0, stream>>>(feats, weight, bias, nidx, out, N);
}